// ProbHypernet_52432960750123
// MI455X (gfx1250) — compile-verified
//
#include <hip/hip_runtime.h>
#include <hip/hip_bf16.h>
#include <math.h>

typedef __attribute__((ext_vector_type(2))) float v2f;
typedef __attribute__((ext_vector_type(8))) float v8f;

#define IN_DIM  1024
#define OUT_DIM 1024
#define H_DIM   256
#define RR      1025   // in_dim + 1 (bias row)
#define MBATCH  128
#define PSPLIT  16     // r-dimension split for the E-stream kernel
#define RCHUNK  65     // ceil(1025 / 16)

// ---------------------------------------------------------------------------
// Branchless f32 WMMA 16x16x4 accumulation loop (K multiple of 4, all
// addresses pre-clamped to valid memory).
//   ap = A + row*lda + 2*half   (per-lane A row, half-wave K offset)
//   bp = B + colB*bn            (per-lane B column base), element k at bp[k*bk]
// A-matrix lane layout (ISA 7.12.2, 32-bit 16x4): lanes 0-15 hold M=0..15
// K={kb,kb+1}; lanes 16-31 hold M=0..15 K={kb+2,kb+3}. B mirrors transposed.
// ---------------------------------------------------------------------------
__device__ __forceinline__ v8f wmma_loop_f32(const float* __restrict__ ap,
                                             const float* __restrict__ bp,
                                             int bk, int half, int K, v8f c) {
#pragma unroll 4
  for (int kb = 0; kb < K; kb += 4) {
    const int k0 = kb + 2 * half;
    v2f a, b;
    a.x = ap[kb];
    a.y = ap[kb + 1];
    b.x = bp[k0 * bk];
    b.y = bp[(k0 + 1) * bk];
    c = __builtin_amdgcn_wmma_f32_16x16x4_f32(false, a, false, b,
                                              (short)0, c, false, false);
  }
  return c;
}

__device__ __forceinline__ float block_reduce_sum(float v, float* sm) {
  const int tid = threadIdx.x;
  sm[tid] = v;
  __syncthreads();
  for (int s = blockDim.x >> 1; s > 0; s >>= 1) {
    if (tid < s) sm[tid] += sm[tid + s];
    __syncthreads();
  }
  float r = sm[0];
  __syncthreads();
  return r;
}

// ---------------------------------------------------------------------------
// msq[r] = sum_c M[r,c]^2   (so sum(Mb*Mb) = sum_r mu^2 * msq, M read once)
// ---------------------------------------------------------------------------
__global__ void k_msq(const float* __restrict__ M, float* __restrict__ msq) {
  __shared__ float sm[256];
  const int r = blockIdx.x;
  const float4 mv = ((const float4*)(M + (size_t)r * OUT_DIM))[threadIdx.x];
  float s = mv.x * mv.x + mv.y * mv.y + mv.z * mv.z + mv.w * mv.w;
  s = block_reduce_sum(s, sm);
  if (threadIdx.x == 0) msq[r] = s;
}

// ---------------------------------------------------------------------------
// h = relu(x @ W_xh^T + b_xh)   [128 x 256]   grid (16,8), block 32
// No guards needed: K=1024 and N=256 are tile-aligned.
// ---------------------------------------------------------------------------
__global__ void k_hidden(const float* __restrict__ x,
                         const float* __restrict__ Wxh,
                         const float* __restrict__ bxh,
                         float* __restrict__ h) {
  const int n0 = blockIdx.x * 16, m0 = blockIdx.y * 16;
  const int lane = threadIdx.x & 31, half = lane >> 4, l = lane & 15;
  const int row = m0 + l, col = n0 + l;
  v8f c = {0.f, 0.f, 0.f, 0.f, 0.f, 0.f, 0.f, 0.f};
  c = wmma_loop_f32(x + row * IN_DIM + 2 * half,      // A row
                    Wxh + col * IN_DIM,               // B col (W^T: bk=1)
                    1, half, IN_DIM, c);
  const float bias = bxh[col];
#pragma unroll
  for (int v = 0; v < 8; ++v) {
    const int orow = m0 + v + 8 * half;
    h[orow * H_DIM + col] = fmaxf(c[v] + bias, 0.f);
  }
}

// ---------------------------------------------------------------------------
// mu / logvar_r / logvar_c = h @ W^T + b   (grid.z selects which)
// grid (65, 8, 3), block 32.  Column tail handled by address clamping: D
// column n depends only on B column n, so junk columns are simply not stored.
// ---------------------------------------------------------------------------
__global__ void k_proj(const float* __restrict__ h,
                       const float* __restrict__ Wmu, const float* __restrict__ bmu,
                       const float* __restrict__ Wlr, const float* __restrict__ blr,
                       const float* __restrict__ Wlc, const float* __restrict__ blc,
                       float* __restrict__ mu, float* __restrict__ lvr,
                       float* __restrict__ lvc) {
  const int which = blockIdx.z;
  const int n0 = blockIdx.x * 16, m0 = blockIdx.y * 16;
  const float *W, *bb;
  float* out;
  int N, ldo;
  if (which == 0)      { W = Wmu; bb = bmu; out = mu;  N = RR;      ldo = RR; }
  else if (which == 1) { W = Wlr; bb = blr; out = lvr; N = RR;      ldo = RR; }
  else                 { W = Wlc; bb = blc; out = lvc; N = OUT_DIM; ldo = OUT_DIM;
                         if (n0 >= OUT_DIM) return; }
  const int lane = threadIdx.x & 31, half = lane >> 4, l = lane & 15;
  const int row = m0 + l;
  const int col = n0 + l;
  const int colB = (col < N) ? col : (N - 1);   // clamp address, no branch
  v8f c = {0.f, 0.f, 0.f, 0.f, 0.f, 0.f, 0.f, 0.f};
  c = wmma_loop_f32(h + row * H_DIM + 2 * half,
                    W + colB * H_DIM,
                    1, half, H_DIM, c);
  if (col < N) {
    const float bias = bb[col];
#pragma unroll
    for (int v = 0; v < 8; ++v) {
      const int orow = m0 + v + 8 * half;
      out[orow * ldo + col] = c[v] + bias;
    }
  }
}

// ---------------------------------------------------------------------------
// Per-sample row stats:
//   a[m,r]  = x_aug * mu            (for a @ M)
//   b[m,r]  = x_aug * exp(lv_r/2)   (for E contraction)
//   sc[m,c] = exp(lv_c/2)
//   scal[m] = { S_vr, S_lvr, Q, S_vc, S_lvc }   for D_KL
// grid 128, block 256
// ---------------------------------------------------------------------------
__global__ void k_rowstats(const float* __restrict__ x,
                           const float* __restrict__ mu,
                           const float* __restrict__ lvr,
                           const float* __restrict__ lvc,
                           const float* __restrict__ msq,
                           float* __restrict__ a, float* __restrict__ b,
                           float* __restrict__ sc, float* __restrict__ scal) {
  __shared__ float sm[256];
  const int m = blockIdx.x, tid = threadIdx.x;
  float s_vr = 0.f, s_lvr = 0.f, q = 0.f;
  for (int r = tid; r < RR; r += 256) {
    const float muv = mu[m * RR + r];
    const float lv  = lvr[m * RR + r];
    const float xa  = (r < IN_DIM) ? x[m * IN_DIM + r] : 1.f;
    a[m * RR + r] = xa * muv;
    b[m * RR + r] = xa * expf(0.5f * lv);
    s_vr  += expf(lv);
    s_lvr += lv;
    q     += muv * muv * msq[r];
  }
  float s_vc = 0.f, s_lvc = 0.f;
  for (int cc = tid; cc < OUT_DIM; cc += 256) {
    const float lv = lvc[m * OUT_DIM + cc];
    sc[m * OUT_DIM + cc] = expf(0.5f * lv);
    s_vc  += expf(lv);
    s_lvc += lv;
  }
  s_vr  = block_reduce_sum(s_vr,  sm);
  s_lvr = block_reduce_sum(s_lvr, sm);
  q     = block_reduce_sum(q,     sm);
  s_vc  = block_reduce_sum(s_vc,  sm);
  s_lvc = block_reduce_sum(s_lvc, sm);
  if (tid == 0) {
    float* p = scal + m * 5;
    p[0] = s_vr; p[1] = s_lvr; p[2] = q; p[3] = s_vc; p[4] = s_lvc;
  }
}

// ---------------------------------------------------------------------------
// AM = a @ M   [128 x 1024]   grid (64,8), block 32
// K = 1025: clean K=1024 WMMA loop + one tail WMMA with selected operands
// (unconditional loads from valid addresses, then v_cndmask zeroing).
// ---------------------------------------------------------------------------
__global__ void k_am(const float* __restrict__ a, const float* __restrict__ M,
                     float* __restrict__ AM) {
  const int n0 = blockIdx.x * 16, m0 = blockIdx.y * 16;
  const int lane = threadIdx.x & 31, half = lane >> 4, l = lane & 15;
  const int row = m0 + l, col = n0 + l;
  v8f c = {0.f, 0.f, 0.f, 0.f, 0.f, 0.f, 0.f, 0.f};
  c = wmma_loop_f32(a + row * RR + 2 * half,           // A row, lda = 1025
                    M + col,                           // B col, bk = 1024
                    OUT_DIM, half, 1024, c);
  // Tail: k = 1024 (only lanes with half==0 carry a valid K pair {1024, x}).
  {
    const float av = a[row * RR + 1024];               // always a valid address
    const float bv = M[1024 * OUT_DIM + col];          // always a valid address
    v2f at, bt;
    at.x = half ? 0.f : av;
    at.y = 0.f;
    bt.x = half ? 0.f : bv;
    bt.y = 0.f;
    c = __builtin_amdgcn_wmma_f32_16x16x4_f32(false, at, false, bt,
                                              (short)0, c, false, false);
  }
#pragma unroll
  for (int v = 0; v < 8; ++v) {
    const int orow = m0 + v + 8 * half;
    AM[orow * OUT_DIM + col] = c[v];
  }
}

// ---------------------------------------------------------------------------
// THE bandwidth kernel: partial[p][m][c] = sum_{r in chunk p} b[m,r]*E[m,r,c]
// E is 537 MB -> streamed exactly once.  grid (128, 16), block 256, each
// thread owns 4 consecutive columns (b128 loads).  Prefetch is speculative
// (ISA 10.5: translation failures silently dropped), so no bounds branch.
// ---------------------------------------------------------------------------
__global__ void k_estream(const float* __restrict__ E,
                          const float* __restrict__ b,
                          float* __restrict__ partial) {
  const int m = blockIdx.x, p = blockIdx.y;
  const int c0 = threadIdx.x * 4;
  const int r0 = p * RCHUNK;
  const int r1 = (r0 + RCHUNK < RR) ? (r0 + RCHUNK) : RR;
  const float* __restrict__ Em = E + (size_t)m * RR * OUT_DIM;
  const float* __restrict__ bm = b + m * RR;
  float4 acc = make_float4(0.f, 0.f, 0.f, 0.f);
#pragma unroll 4
  for (int r = r0; r < r1; ++r) {
    __builtin_prefetch(Em + (size_t)(r + 8) * OUT_DIM + c0, 0, 3);
    const float bv = bm[r];                            // uniform -> scalar load
    const float4 ev = *(const float4*)(Em + (size_t)r * OUT_DIM + c0);
    acc.x = fmaf(bv, ev.x, acc.x);
    acc.y = fmaf(bv, ev.y, acc.y);
    acc.z = fmaf(bv, ev.z, acc.z);
    acc.w = fmaf(bv, ev.w, acc.w);
  }
  *(float4*)(partial + ((size_t)p * MBATCH + m) * OUT_DIM + c0) = acc;
}

// ---------------------------------------------------------------------------
// out[m,c] = AM[m,c] + sc[m,c] * sum_p partial[p][m][c]   grid 128, block 256
// ---------------------------------------------------------------------------
__global__ void k_finalize(const float* __restrict__ AM,
                           const float* __restrict__ sc,
                           const float* __restrict__ partial,
                           float* __restrict__ out) {
  const int m = blockIdx.x;
  const int c0 = threadIdx.x * 4;
  float4 s = make_float4(0.f, 0.f, 0.f, 0.f);
#pragma unroll
  for (int p = 0; p < PSPLIT; ++p) {
    const float4 pv =
        *(const float4*)(partial + ((size_t)p * MBATCH + m) * OUT_DIM + c0);
    s.x += pv.x; s.y += pv.y; s.z += pv.z; s.w += pv.w;
  }
  const float4 am  = *(const float4*)(AM + (size_t)m * OUT_DIM + c0);
  const float4 scv = *(const float4*)(sc + (size_t)m * OUT_DIM + c0);
  float4 o;
  o.x = fmaf(scv.x, s.x, am.x);
  o.y = fmaf(scv.y, s.y, am.y);
  o.z = fmaf(scv.z, s.z, am.z);
  o.w = fmaf(scv.w, s.w, am.w);
  *(float4*)(out + (size_t)m * OUT_DIM + c0) = o;
}

// ---------------------------------------------------------------------------
// D_KL = mean_m 0.5*(S_vr*S_vc + Q - r*c - c*S_lvr - r*S_lvc)
// grid 1, block 128
// ---------------------------------------------------------------------------
__global__ void k_dkl(const float* __restrict__ scal, float* __restrict__ out) {
  __shared__ float sm[128];
  const int m = threadIdx.x;
  const float* p = scal + m * 5;
  const float term =
      0.5f * (p[0] * p[3] + p[2] - (float)((long)RR * OUT_DIM)
              - (float)OUT_DIM * p[1] - (float)RR * p[4]);
  sm[m] = term;
  __syncthreads();
  for (int s = 64; s > 0; s >>= 1) {
    if (m < s) sm[m] += sm[m + s];
    __syncthreads();
  }
  if (m == 0) out[0] = sm[0] * (1.f / (float)MBATCH);
}

// ---------------------------------------------------------------------------
// Workspace layout (floats)
// ---------------------------------------------------------------------------
#define OFF_H     0          //  32768
#define OFF_MU    32768      // 131200
#define OFF_LVR   163968     // 131200
#define OFF_LVC   295168     // 131072
#define OFF_A     426240     // 131200
#define OFF_B     557440     // 131200
#define OFF_SC    688640     // 131072
#define OFF_MSQ   819712     //   1056
#define OFF_SCAL  820768     //    640
#define OFF_AM    821408     // 131072
#define OFF_PART  952480     // 2097152 -> total 3,049,632 floats (~12.2 MB)

extern "C" void kernel_launch(void* const* d_in, const int* in_sizes, int n_in,
                              void* d_out, int out_size, void* d_ws, size_t ws_size,
                              hipStream_t stream) {
  const float* x    = (const float*)d_in[0];
  const float* E    = (const float*)d_in[1];
  const float* M    = (const float*)d_in[2];
  const float* Wxh  = (const float*)d_in[3];
  const float* bxh  = (const float*)d_in[4];
  const float* Wmu  = (const float*)d_in[5];
  const float* bmu  = (const float*)d_in[6];
  const float* Wlr  = (const float*)d_in[7];
  const float* blr  = (const float*)d_in[8];
  const float* Wlc  = (const float*)d_in[9];
  const float* blc  = (const float*)d_in[10];

  float* ws   = (float*)d_ws;
  float* h    = ws + OFF_H;
  float* mu   = ws + OFF_MU;
  float* lvr  = ws + OFF_LVR;
  float* lvc  = ws + OFF_LVC;
  float* a    = ws + OFF_A;
  float* b    = ws + OFF_B;
  float* sc   = ws + OFF_SC;
  float* msq  = ws + OFF_MSQ;
  float* scal = ws + OFF_SCAL;
  float* AM   = ws + OFF_AM;
  float* part = ws + OFF_PART;
  float* out  = (float*)d_out;

  k_msq<<<RR, 256, 0, stream>>>(M, msq);
  k_hidden<<<dim3(H_DIM / 16, MBATCH / 16), 32, 0, stream>>>(x, Wxh, bxh, h);
  k_proj<<<dim3(65, MBATCH / 16, 3), 32, 0, stream>>>(h, Wmu, bmu, Wlr, blr,
                                                      Wlc, blc, mu, lvr, lvc);
  k_rowstats<<<MBATCH, 256, 0, stream>>>(x, mu, lvr, lvc, msq, a, b, sc, scal);
  k_am<<<dim3(OUT_DIM / 16, MBATCH / 16), 32, 0, stream>>>(a, M, AM);
  k_estream<<<dim3(MBATCH, PSPLIT), 256, 0, stream>>>(E, b, part);
  k_finalize<<<MBATCH, 256, 0, stream>>>(AM, sc, part, out);
  k_dkl<<<1, 128, 0, stream>>>(scal, out + (size_t)MBATCH * OUT_DIM);
}